// embedding_net_30099130811081
// MI455X (gfx1250) — compile-verified
//
#include <hip/hip_runtime.h>

typedef __attribute__((ext_vector_type(2))) float v2f;
typedef __attribute__((ext_vector_type(4))) float v4f;
typedef __attribute__((ext_vector_type(8))) float v8f;

#define EMB_STRIDE 132   // padded row stride (floats) for transposed embed [w][m]
#define RED_STRIDE 132   // padded row stride (floats) for reducer [r][m]

// out[n,r,w] = sum_m reducer[n,r,m] * Horner(coeff[idx(x[n,m])], x0)[w]
__global__ __launch_bounds__(128) void embed_reduce_wmma(
    const float* __restrict__ x,        // [4096,128,1]
    const float* __restrict__ coeff,    // [1024,6,64]
    const float* __restrict__ reducer,  // [4096,16,128]
    float* __restrict__ out)            // [4096,16,64]
{
    __shared__ float s_emb[64 * EMB_STRIDE];  // embed transposed: [w][m], 33 KB
    __shared__ float s_red[16 * RED_STRIDE];  // reducer[n]:       [r][m],  8.4 KB

    const int n    = blockIdx.x;
    const int tid  = threadIdx.x;       // 0..127
    const int lane = tid & 31;
    const int wave = tid >> 5;          // 0..3  -> W-tile

    // ---- Stage 0: stage reducer[n] (16x128 f32) into LDS ----
    {
        const int r  = tid >> 3;          // 0..15
        const int c0 = (tid & 7) * 16;    // 0,16,...,112
        const float* src = reducer + ((size_t)n * 16 + r) * 128 + c0;
        float*       dst = s_red + r * RED_STRIDE + c0;
        #pragma unroll
        for (int j = 0; j < 4; ++j)
            *(v4f*)(dst + 4 * j) = *(const v4f*)(src + 4 * j);
    }

    // ---- Stage 1: per-m degree-5 Horner embedding, store transposed [w][m] ----
    {
        const int   m         = tid;                    // 0..127
        const float xv        = x[(size_t)n * 128 + m]; // SRMIN == 0
        const float inv_delta = 128.0f;                 // NG / (SRMAX-SRMIN) = 1024/8
        const float delta     = 0.0078125f;             // 8/1024
        const int   idx       = (int)(xv * inv_delta);
        const float x0        = xv - (float)idx * delta;
        const float* crow     = coeff + (size_t)idx * (6 * 64);

        #pragma unroll
        for (int w4 = 0; w4 < 16; ++w4) {
            v4f e = *(const v4f*)(crow + 0 * 64 + 4 * w4);
            #pragma unroll
            for (int i = 1; i < 6; ++i) {
                v4f c = *(const v4f*)(crow + i * 64 + 4 * w4);
                e = e * x0 + c;  // fma per component
            }
            float* d = s_emb + (4 * w4) * EMB_STRIDE + m;  // conflict-free: bank = 4w+m
            d[0 * EMB_STRIDE] = e.x;
            d[1 * EMB_STRIDE] = e.y;
            d[2 * EMB_STRIDE] = e.z;
            d[3 * EMB_STRIDE] = e.w;
        }
    }

    __syncthreads();

    // ---- Stage 2: (16x128) @ (128x64) via V_WMMA_F32_16X16X4_F32 ----
    // A 16x4 layout: lanes 0-15 -> M=lane, v0=K0,v1=K1; lanes 16-31 -> v0=K2,v1=K3
    // B 4x16 layout: lanes 0-15 -> N=lane, v0=K0,v1=K1; lanes 16-31 -> v0=K2,v1=K3
    const int w0   = wave * 16;     // W-tile base
    const int half = lane >> 4;     // 0 or 1 -> K pair (0,1) vs (2,3)
    const int l16  = lane & 15;

    const float* aptr = s_red + l16 * RED_STRIDE + 2 * half;          // reducer[r=l16][...]
    const float* bptr = s_emb + (w0 + l16) * EMB_STRIDE + 2 * half;   // embed^T[w][...]

    v8f acc = {};
    #pragma unroll
    for (int k = 0; k < 32; ++k) {   // K-tile: m = 4k .. 4k+3
        v2f a = *(const v2f*)(aptr + 4 * k);
        v2f b = *(const v2f*)(bptr + 4 * k);
        // 8 args: (neg_a, A, neg_b, B, c_mod, C, reuse_a, reuse_b)
        acc = __builtin_amdgcn_wmma_f32_16x16x4_f32(
            false, a, false, b, (short)0, acc, false, false);
    }

    // ---- Stage 3: write D fragment (VGPR i -> row i + 8*half, col w0+l16) ----
    float* optr = out + (size_t)n * (16 * 64) + (half * 8) * 64 + w0 + l16;
    #pragma unroll
    for (int i = 0; i < 8; ++i)
        optr[i * 64] = acc[i];
}

extern "C" void kernel_launch(void* const* d_in, const int* in_sizes, int n_in,
                              void* d_out, int out_size, void* d_ws, size_t ws_size,
                              hipStream_t stream) {
    const float* x       = (const float*)d_in[0];   // 4096*128
    const float* coeff   = (const float*)d_in[1];   // 1024*6*64
    const float* reducer = (const float*)d_in[2];   // 4096*16*128
    float*       out     = (float*)d_out;           // 4096*16*64

    const int N = in_sizes[0] / 128;                // 4096 blocks, one per n
    embed_reduce_wmma<<<N, 128, 0, stream>>>(x, coeff, reducer, out);
}